// GCN3_46617575031251
// MI455X (gfx1250) — compile-verified
//
#include <hip/hip_runtime.h>
#include <hip/hip_bf16.h>

typedef __attribute__((ext_vector_type(2))) float v2f;
typedef __attribute__((ext_vector_type(8))) float v8f;

#define NNODES 50000
#define NEDGES 800000
#define F_IN 128
#define H_DIM 96
#define C_DIM 32

// ---------------------------------------------------------------- utilities

__global__ void fill_kernel(float* __restrict__ p, int n, float v) {
    int i = blockIdx.x * blockDim.x + threadIdx.x;
    if (i < n) p[i] = v;
}

__global__ void deg_kernel(const int* __restrict__ dst, float* __restrict__ deg, int e) {
    int i = blockIdx.x * blockDim.x + threadIdx.x;
    if (i < e) atomicAdd(&deg[dst[i]], 1.0f);
}

__global__ void dinv_kernel(float* __restrict__ d, int n) {
    int i = blockIdx.x * blockDim.x + threadIdx.x;
    if (i < n) d[i] = rsqrtf(d[i]);   // deg >= 1 always (self-loop)
}

__global__ void norm_kernel(const int* __restrict__ src, const int* __restrict__ dst,
                            const float* __restrict__ dinv, float* __restrict__ norm, int e) {
    int i = blockIdx.x * blockDim.x + threadIdx.x;
    if (i < e) norm[i] = dinv[src[i]] * dinv[dst[i]];
}

// ------------------------------------------------------- WMMA f32 GEMM
// Out[M x F] = A[M x K] @ W[K x F].  One wave32 per 16x16 output tile,
// K-loop in steps of 4 using V_WMMA_F32_16X16X4_F32.
// A frag (16x4): lanes 0-15 hold M=lane, {K=k0,k0+1}; lanes 16-31 hold {K=k0+2,k0+3}.
// B frag (4x16): lanes 0-15 hold N=lane, {K=k0,k0+1}; lanes 16-31 hold {K=k0+2,k0+3}.
// C/D: VGPR v, lanes 0-15 -> (M=v, N=lane); lanes 16-31 -> (M=v+8, N=lane-16).
__global__ void gemm_wmma_kernel(const float* __restrict__ A, const float* __restrict__ W,
                                 float* __restrict__ Out, int M, int K, int F) {
    int wave = (blockIdx.x * blockDim.x + threadIdx.x) >> 5;
    int lane = threadIdx.x & 31;
    int ntn  = F >> 4;                 // tiles along F
    int ntm  = M >> 4;                 // tiles along M
    if (wave >= ntm * ntn) return;
    int tm = wave / ntn;
    int tn = wave - tm * ntn;

    int lo   = lane & 15;
    int koff = (lane >> 4) << 1;       // 0 or 2
    int arow = tm * 16 + lo;           // A row for this lane
    int bcol = tn * 16 + lo;           // B col for this lane

    const float* Arow = A + (long)arow * K;
    v8f acc = {};
    #pragma unroll 4
    for (int k = 0; k < K; k += 4) {
        v2f a, b;
        a.x = Arow[k + koff];
        a.y = Arow[k + koff + 1];
        b.x = W[(k + koff) * F + bcol];
        b.y = W[(k + koff + 1) * F + bcol];
        acc = __builtin_amdgcn_wmma_f32_16x16x4_f32(
            /*neg_a=*/false, a, /*neg_b=*/false, b,
            /*c_mod=*/(short)0, acc, /*reuse_a=*/false, /*reuse_b=*/false);
    }

    int orow = tm * 16 + ((lane >> 4) << 3);   // +0 or +8
    int ocol = tn * 16 + lo;
    #pragma unroll
    for (int v = 0; v < 8; ++v)
        Out[(long)(orow + v) * F + ocol] = acc[v];
}

// ------------------------------------------------------- edge scatter-add
// out[dst[e]][f] += norm[e] * xw[src[e]][f]   (one thread per (edge, feature))
__global__ void scatter_kernel(const float* __restrict__ xw, const float* __restrict__ norm,
                               const int* __restrict__ src, const int* __restrict__ dst,
                               float* __restrict__ out, int e, int F) {
    long idx = (long)blockIdx.x * blockDim.x + threadIdx.x;
    long total = (long)e * F;
    if (idx >= total) return;
    int ed = (int)(idx / F);
    int f  = (int)(idx - (long)ed * F);
    int s  = src[ed], d = dst[ed];
    float v = norm[ed] * xw[(long)s * F + f];
    atomicAdd(&out[(long)d * F + f], v);
}

// ------------------------------------------------------- finalize (self-loop + bias + relu)
// out[n][f] = relu(out[n][f] + dinv[n]^2 * xw[n][f] + b[f])   (in place)
__global__ void finalize_relu_kernel(float* __restrict__ out, const float* __restrict__ xw,
                                     const float* __restrict__ dinv, const float* __restrict__ b,
                                     int n, int F) {
    long idx = (long)blockIdx.x * blockDim.x + threadIdx.x;
    long total = (long)n * F;
    if (idx >= total) return;
    int node = (int)(idx / F);
    int f    = (int)(idx - (long)node * F);
    float di = dinv[node];
    float v  = out[idx] + di * di * xw[idx] + b[f];
    out[idx] = v > 0.0f ? v : 0.0f;
}

// ------------------------------------------------------- final layer: self-loop + bias + log_softmax
// One wave32 per node row (C=32 classes). agg is d_out (already scatter-accumulated).
__global__ void logsoftmax_kernel(float* __restrict__ out, const float* __restrict__ xw,
                                  const float* __restrict__ dinv, const float* __restrict__ b,
                                  int n) {
    int wavesPerBlock = blockDim.x >> 5;
    int node = blockIdx.x * wavesPerBlock + (threadIdx.x >> 5);
    int lane = threadIdx.x & 31;
    if (node >= n) return;
    long idx = (long)node * C_DIM + lane;
    float di = dinv[node];
    float v  = out[idx] + di * di * xw[idx] + b[lane];

    float m = v;
    #pragma unroll
    for (int off = 16; off > 0; off >>= 1)
        m = fmaxf(m, __shfl_xor(m, off, 32));
    float ex = __expf(v - m);
    float s = ex;
    #pragma unroll
    for (int off = 16; off > 0; off >>= 1)
        s += __shfl_xor(s, off, 32);
    out[idx] = v - m - __logf(s);
}

// ---------------------------------------------------------------- launcher

static inline int cdiv(long a, int b) { return (int)((a + b - 1) / b); }

extern "C" void kernel_launch(void* const* d_in, const int* in_sizes, int n_in,
                              void* d_out, int out_size, void* d_ws, size_t ws_size,
                              hipStream_t stream) {
    const float* x  = (const float*)d_in[0];
    const int*   ei = (const int*)d_in[1];
    const float* W1 = (const float*)d_in[2];
    const float* b1 = (const float*)d_in[3];
    const float* W2 = (const float*)d_in[4];
    const float* b2 = (const float*)d_in[5];
    const float* W3 = (const float*)d_in[6];
    const float* b3 = (const float*)d_in[7];
    float* out = (float*)d_out;

    const int N = in_sizes[0] / F_IN;     // 50000
    const int E = in_sizes[1] / 2;        // 800000
    const int* src = ei;
    const int* dst = ei + E;

    // workspace layout (floats)
    float* norm = (float*)d_ws;           // E
    float* dinv = norm + E;               // N
    float* bufA = dinv + N;               // N * 96  (holds XW for each layer)
    float* bufB = bufA + (long)N * H_DIM; // N * 96  (h1)
    float* bufC = bufB + (long)N * H_DIM; // N * 96  (h2)

    const int T = 256;

    // degrees -> dinv -> per-edge norm
    fill_kernel<<<cdiv(N, T), T, 0, stream>>>(dinv, N, 1.0f);            // self-loop
    deg_kernel<<<cdiv(E, T), T, 0, stream>>>(dst, dinv, E);
    dinv_kernel<<<cdiv(N, T), T, 0, stream>>>(dinv, N);
    norm_kernel<<<cdiv(E, T), T, 0, stream>>>(src, dst, dinv, norm, E);

    // ---- layer 1: x(128) @ W1 -> 96, aggregate, +b1, relu -> bufB
    {
        int waves = (N / 16) * (H_DIM / 16);
        gemm_wmma_kernel<<<cdiv((long)waves * 32, T), T, 0, stream>>>(x, W1, bufA, N, F_IN, H_DIM);
        fill_kernel<<<cdiv((long)N * H_DIM, T), T, 0, stream>>>(bufB, N * H_DIM, 0.0f);
        scatter_kernel<<<cdiv((long)E * H_DIM, T), T, 0, stream>>>(bufA, norm, src, dst, bufB, E, H_DIM);
        finalize_relu_kernel<<<cdiv((long)N * H_DIM, T), T, 0, stream>>>(bufB, bufA, dinv, b1, N, H_DIM);
    }

    // ---- layer 2: h1(96) @ W2 -> 96, aggregate, +b2, relu -> bufC
    {
        int waves = (N / 16) * (H_DIM / 16);
        gemm_wmma_kernel<<<cdiv((long)waves * 32, T), T, 0, stream>>>(bufB, W2, bufA, N, H_DIM, H_DIM);
        fill_kernel<<<cdiv((long)N * H_DIM, T), T, 0, stream>>>(bufC, N * H_DIM, 0.0f);
        scatter_kernel<<<cdiv((long)E * H_DIM, T), T, 0, stream>>>(bufA, norm, src, dst, bufC, E, H_DIM);
        finalize_relu_kernel<<<cdiv((long)N * H_DIM, T), T, 0, stream>>>(bufC, bufA, dinv, b2, N, H_DIM);
    }

    // ---- layer 3: h2(96) @ W3 -> 32, aggregate into d_out, +b3, log_softmax
    {
        int waves = (N / 16) * (C_DIM / 16);
        gemm_wmma_kernel<<<cdiv((long)waves * 32, T), T, 0, stream>>>(bufC, W3, bufA, N, H_DIM, C_DIM);
        fill_kernel<<<cdiv((long)N * C_DIM, T), T, 0, stream>>>(out, N * C_DIM, 0.0f);
        scatter_kernel<<<cdiv((long)E * C_DIM, T), T, 0, stream>>>(bufA, norm, src, dst, out, E, C_DIM);
        logsoftmax_kernel<<<cdiv((long)N * 32, T), T, 0, stream>>>(out, bufA, dinv, b3, N);
    }
}